// DDI_gcn_85667417686478
// MI455X (gfx1250) — compile-verified
//
#include <hip/hip_runtime.h>
#include <stdint.h>

#define MEDNUM  50000
#define FEATDIM 128
#define NTOT    (2 * MEDNUM)
#define WPB     8      // waves per 256-thread block (wave32)
#define SCAN_T  1024
#define ELEMS   8      // elements per thread in the scan

static_assert(FEATDIM == 128, "layout assumes 128 features = 32 lanes x float4");

// ---------------- K1: zero the per-row counts ----------------
__global__ void ddi_zero_counts(int* __restrict__ counts, int n) {
  int i = blockIdx.x * blockDim.x + threadIdx.x;
  if (i < n) counts[i] = 0;
}

// ---------------- K2: histogram rows ----------------
__global__ void ddi_count_rows(const int* __restrict__ row_idx,
                               int* __restrict__ counts, int nnz) {
  int e = blockIdx.x * blockDim.x + threadIdx.x;
  if (e < nnz) atomicAdd(&counts[row_idx[e]], 1);
}

// ---------------- K3: exclusive scan (1 block; shfl-based, few barriers) ----
__global__ __launch_bounds__(SCAN_T)
void ddi_scan(const int* __restrict__ counts, int* __restrict__ offsets,
              int* __restrict__ cursor, int n) {
  __shared__ int swave[32];
  __shared__ int stotal;
  __shared__ int sbase;
  const int tid  = threadIdx.x;
  const int lane = tid & 31;
  const int wid  = tid >> 5;
  if (tid == 0) sbase = 0;
  __syncthreads();

  for (int base = 0; base < n; base += SCAN_T * ELEMS) {
    const int i0 = base + tid * ELEMS;
    int vk[ELEMS];
    int s = 0;
#pragma unroll
    for (int k = 0; k < ELEMS; ++k) {
      const int i = i0 + k;
      vk[k] = (i < n) ? counts[i] : 0;
      s += vk[k];
    }
    // wave32 inclusive scan of per-thread totals
    int ws_ = s;
#pragma unroll
    for (int d = 1; d < 32; d <<= 1) {
      int t = __shfl_up(ws_, d, 32);
      if (lane >= d) ws_ += t;
    }
    if (lane == 31) swave[wid] = ws_;
    __syncthreads();                       // B1
    if (wid == 0) {
      int wv = swave[lane];
      int t2 = wv;
#pragma unroll
      for (int d = 1; d < 32; d <<= 1) {
        int t = __shfl_up(t2, d, 32);
        if (lane >= d) t2 += t;
      }
      swave[lane] = t2 - wv;               // exclusive wave offset
      if (lane == 31) stotal = t2;         // chunk total
    }
    __syncthreads();                       // B2
    int run = sbase + swave[wid] + (ws_ - s);  // exclusive prefix for this thread
#pragma unroll
    for (int k = 0; k < ELEMS; ++k) {
      const int i = i0 + k;
      if (i < n) {
        offsets[i] = run;
        cursor[i]  = run;
      }
      run += vk[k];
    }
    __syncthreads();                       // B3
    if (tid == 0) sbase += stotal;
    __syncthreads();                       // B4
  }
  if (tid == 0) offsets[n] = sbase;
}

// ---------------- K4: bin edges into packed (val,col) records ----------------
__global__ void ddi_fill_bins(const float* __restrict__ vals,
                              const int* __restrict__ row_idx,
                              const int* __restrict__ col_idx,
                              int* __restrict__ cursor,
                              int2* __restrict__ pedge, int nnz) {
  int e = blockIdx.x * blockDim.x + threadIdx.x;
  if (e >= nnz) return;
  const int r   = row_idx[e];
  const int pos = atomicAdd(&cursor[r], 1);
  pedge[pos] = make_int2(__float_as_int(vals[e]), col_idx[e]);  // single b64 store
}

// ---------------- K5: fused wave-per-row aggregation + finalize ----------------
// Wave owns output row `row`: accumulates segment sums for rows `row` and
// `row+MEDNUM`, applies relu and the inter blend, writes 512B to out.
// Edge windows are staged into LDS with gfx1250 async-to-LDS b64 loads
// (ASYNCcnt path) and broadcast-read by all 32 lanes.
__global__ __launch_bounds__(256)
void ddi_aggregate_combine(const int* __restrict__ offsets,
                           const int2* __restrict__ pedge,
                           const float* __restrict__ mEmbed,
                           const float* __restrict__ inter,
                           float* __restrict__ out) {
  __shared__ __align__(16) int2 s_e[WPB][32];

  const int lane = threadIdx.x & 31;
  const int wid  = threadIdx.x >> 5;
  const int row  = blockIdx.x * WPB + wid;
  if (row >= MEDNUM) return;
  const float w = inter[0];

  // Hoist both halves' row descriptors so their latency overlaps the first
  // async window.
  const int beg0 = offsets[row];
  const int end0 = offsets[row + 1];
  const int beg1 = offsets[row + MEDNUM];
  const int end1 = offsets[row + MEDNUM + 1];

  float4 res[2];
#pragma unroll
  for (int h = 0; h < 2; ++h) {
    const int beg = h ? beg1 : beg0;
    const int end = h ? end1 : end0;
    float4 acc = make_float4(0.f, 0.f, 0.f, 0.f);

    for (int k = beg; k < end; k += 32) {
      const int kk = k + lane;
      // WAR guard: the async engine writes LDS outside the wave's DS pipeline;
      // make sure prior broadcast reads of this window have drained.
      asm volatile("s_wait_dscnt 0x0" ::: "memory");
      if (kk < end) {
        unsigned d = (unsigned)(unsigned long long)&s_e[wid][lane];
        unsigned long long a = (unsigned long long)(pedge + kk);
        asm volatile("global_load_async_to_lds_b64 %0, %1, off"
                     :: "v"(d), "v"(a) : "memory");
      }
      asm volatile("s_wait_asynccnt 0x0" ::: "memory");

      const int cnt = min(32, end - k);
      // Unroll so the scheduler can keep 4 broadcast reads + 4 512B gathers
      // in flight before the FMAs consume them (MLP within one wave).
#pragma unroll 4
      for (int j = 0; j < cnt; ++j) {
        const int2  e2 = s_e[wid][j];              // ds_load_b64 broadcast
        const float vj = __int_as_float(e2.x);
        const int   cj = e2.y;
        const int   cm = cj - (cj >= MEDNUM ? MEDNUM : 0);
        const float4 em =
            *(const float4*)(mEmbed + (size_t)cm * FEATDIM + 4 * lane);
        acc.x += vj * em.x;
        acc.y += vj * em.y;
        acc.z += vj * em.z;
        acc.w += vj * em.w;
      }
    }
    res[h] = acc;
  }

  float4 o;
  o.x = 2.f * (w * fmaxf(res[0].x, 0.f) + (1.f - w) * fmaxf(res[1].x, 0.f));
  o.y = 2.f * (w * fmaxf(res[0].y, 0.f) + (1.f - w) * fmaxf(res[1].y, 0.f));
  o.z = 2.f * (w * fmaxf(res[0].z, 0.f) + (1.f - w) * fmaxf(res[1].z, 0.f));
  o.w = 2.f * (w * fmaxf(res[0].w, 0.f) + (1.f - w) * fmaxf(res[1].w, 0.f));
  *(float4*)(out + (size_t)row * FEATDIM + 4 * lane) = o;
}

extern "C" void kernel_launch(void* const* d_in, const int* in_sizes, int n_in,
                              void* d_out, int out_size, void* d_ws, size_t ws_size,
                              hipStream_t stream) {
  const float* vals    = (const float*)d_in[0];
  const float* mEmbed  = (const float*)d_in[1];
  const float* inter   = (const float*)d_in[2];
  const int*   row_idx = (const int*)d_in[3];
  const int*   col_idx = (const int*)d_in[4];
  float*       out     = (float*)d_out;
  const int nnz = in_sizes[0];

  // Workspace layout (4-byte units; pedge ends up 8B-aligned):
  int*  counts = (int*)d_ws;            // NTOT
  int*  offs   = counts + NTOT;         // NTOT+1 (+1 pad for alignment)
  int*  cursor = offs + (NTOT + 2);     // NTOT
  int2* pedge  = (int2*)(cursor + NTOT);// nnz records

  const int T = 256;

  ddi_zero_counts<<<(NTOT + T - 1) / T, T, 0, stream>>>(counts, NTOT);
  ddi_count_rows<<<(nnz + T - 1) / T, T, 0, stream>>>(row_idx, counts, nnz);
  ddi_scan<<<1, SCAN_T, 0, stream>>>(counts, offs, cursor, NTOT);
  ddi_fill_bins<<<(nnz + T - 1) / T, T, 0, stream>>>(vals, row_idx, col_idx,
                                                     cursor, pedge, nnz);
  ddi_aggregate_combine<<<(MEDNUM + WPB - 1) / WPB, T, 0, stream>>>(
      offs, pedge, mEmbed, inter, out);
}